// GNN_12695923327103
// MI455X (gfx1250) — compile-verified
//
#include <hip/hip_runtime.h>
#include <math.h>

typedef __attribute__((ext_vector_type(2))) float v2f;
typedef __attribute__((ext_vector_type(8))) float v8f;

#define N_GRAPHS 128

// ---------------------------------------------------------------------------
// Small utility kernels
// ---------------------------------------------------------------------------
__global__ void k_fill(float* __restrict__ p, float v, int n) {
    int i = blockIdx.x * blockDim.x + threadIdx.x;
    if (i < n) p[i] = v;
}

__global__ void k_deg_edges(const int* __restrict__ dst, float* __restrict__ deg, int E) {
    int e = blockIdx.x * blockDim.x + threadIdx.x;
    if (e < E) atomicAdd(&deg[dst[e]], 1.0f);
}

__global__ void k_inv_sqrt(const float* __restrict__ deg, float* __restrict__ inv, int n) {
    int i = blockIdx.x * blockDim.x + threadIdx.x;
    if (i < n) inv[i] = rsqrtf(deg[i]);  // deg >= 1 always (self loop)
}

// ---------------------------------------------------------------------------
// WMMA fp32 GEMM:  H[M x NOUT] = X[M x K] * W[K x NOUT]
// One wave computes one 16x16 output tile via V_WMMA_F32_16X16X4_F32.
// blockIdx.x = M-tile; wave id within block = N-tile (blockDim = NTILES*32).
// A tile (16 x K fp32) staged in LDS with a coalesced cooperative fill.
// A layout (32-bit A 16x4): lane%16 = row M, lane/16 selects K-pair {0,1}/{2,3}.
// B layout (4x16): lane%16 = col N, lane/16 selects K row pair.
// C/D layout: vgpr v -> row v + (lane/16)*8, col = lane%16.
// ---------------------------------------------------------------------------
template <int K, int NOUT>
__global__ void k_gemm_wmma(const float* __restrict__ X, const float* __restrict__ W,
                            float* __restrict__ H, int M) {
    constexpr int NTILES = NOUT / 16;
    __shared__ float As[16 * K];

    const int m0  = blockIdx.x * 16;
    const int tid = threadIdx.x;

    // Cooperative, coalesced stage of the 16 x K A-tile into LDS (zero-padded).
    for (int i = tid; i < 16 * K; i += NTILES * 32) {
        int r = i / K, c = i - r * K;
        int gr = m0 + r;
        As[i] = (gr < M) ? X[(size_t)gr * K + c] : 0.0f;
    }
    __syncthreads();

    const int wave = tid >> 5;
    const int lane = tid & 31;
    const int col  = lane & 15;   // N within tile (B/C/D), M row for A
    const int half = lane >> 4;   // selects K-pair / upper C rows
    const int n0   = wave * 16;

    v8f acc = {};
#pragma unroll
    for (int k = 0; k < K; k += 4) {
        const int ka = k + half * 2;
        v2f a, b;
        a.x = As[col * K + ka];
        a.y = As[col * K + ka + 1];
        b.x = W[(size_t)ka       * NOUT + n0 + col];
        b.y = W[(size_t)(ka + 1) * NOUT + n0 + col];
        acc = __builtin_amdgcn_wmma_f32_16x16x4_f32(
            /*neg_a=*/false, a, /*neg_b=*/false, b,
            /*c_mod=*/(short)0, acc, /*reuse_a=*/false, /*reuse_b=*/false);
    }

#pragma unroll
    for (int v = 0; v < 8; ++v) {
        int gr = m0 + v + half * 8;
        if (gr < M) H[(size_t)gr * NOUT + n0 + col] = acc[v];
    }
}

// ---------------------------------------------------------------------------
// GCN aggregation
// ---------------------------------------------------------------------------
// out[i][f] = h[i][f] * inv[i]^2   (self-loop term, also fully initializes out)
template <int F>
__global__ void k_self_init(const float* __restrict__ h, const float* __restrict__ inv,
                            float* __restrict__ out, int n) {
    size_t i = (size_t)blockIdx.x * blockDim.x + threadIdx.x;
    if (i < (size_t)n * F) {
        int node = (int)(i / F);
        float c = inv[node];
        out[i] = h[i] * c * c;
    }
}

// out[dst] += h[src] * inv[src]*inv[dst]; F/4 threads per edge, float4 per lane.
// For F=128 a full wave covers one edge with a 512B coalesced row gather.
template <int F>
__global__ void k_edge_scatter(const int* __restrict__ src, const int* __restrict__ dst,
                               const float* __restrict__ h, const float* __restrict__ inv,
                               float* __restrict__ out, int E) {
    constexpr int TPE = F / 4;
    int t = blockIdx.x * blockDim.x + threadIdx.x;
    int e = t / TPE;
    if (e >= E) return;
    int fo = (t - e * TPE) * 4;
    int s = src[e], d = dst[e];
    float c = inv[s] * inv[d];
    const float4 v = *reinterpret_cast<const float4*>(h + (size_t)s * F + fo);
    float* o = out + (size_t)d * F + fo;
    atomicAdd(o + 0, v.x * c);
    atomicAdd(o + 1, v.y * c);
    atomicAdd(o + 2, v.z * c);
    atomicAdd(o + 3, v.w * c);
}

template <int F>
__global__ void k_bias_relu(float* __restrict__ out, const float* __restrict__ b, size_t total) {
    size_t i = (size_t)blockIdx.x * blockDim.x + threadIdx.x;
    if (i < total) {
        float v = out[i] + b[i & (F - 1)];
        out[i] = v > 0.0f ? v : 0.0f;
    }
}

// ---------------------------------------------------------------------------
// Mean pooling over sorted batch ids: segmented accumulation, atomics only on
// graph-id change (batch is sorted, so ~128 flushes total per feature).
// Block = 64 threads (one per feature); each block walks CHUNK nodes.
// ---------------------------------------------------------------------------
__global__ void k_pool(const float* __restrict__ h, const int* __restrict__ batch,
                       float* __restrict__ sums, float* __restrict__ cnts, int n) {
    constexpr int CHUNK = 512;
    const int f = threadIdx.x;          // 0..63
    int start = blockIdx.x * CHUNK;
    int end   = start + CHUNK;
    if (end > n) end = n;
    if (start >= end) return;

    int cur = batch[start];
    float acc = 0.0f, cnt = 0.0f;
    for (int i = start; i < end; ++i) {
        int g = batch[i];
        if (g != cur) {
            atomicAdd(&sums[cur * 64 + f], acc);
            if (f == 0) atomicAdd(&cnts[cur], cnt);
            acc = 0.0f; cnt = 0.0f; cur = g;
        }
        acc += h[(size_t)i * 64 + f];
        cnt += 1.0f;
    }
    atomicAdd(&sums[cur * 64 + f], acc);
    if (f == 0) atomicAdd(&cnts[cur], cnt);
}

// ---------------------------------------------------------------------------
// Head: pooled = sums/max(cnt,1); logits = pooled @ Wfc + bfc; softmax.
// One thread per graph (128 threads, 1 block) — trivially small.
// ---------------------------------------------------------------------------
__global__ void k_head(const float* __restrict__ sums, const float* __restrict__ cnts,
                       const float* __restrict__ Wfc, const float* __restrict__ bfc,
                       float* __restrict__ out) {
    int g = threadIdx.x;
    if (g >= N_GRAPHS) return;
    float inv_cnt = 1.0f / fmaxf(cnts[g], 1.0f);
    float logits[10];
#pragma unroll
    for (int c = 0; c < 10; ++c) logits[c] = bfc[c];
    for (int f = 0; f < 64; ++f) {
        float p = sums[g * 64 + f] * inv_cnt;
#pragma unroll
        for (int c = 0; c < 10; ++c) logits[c] += p * Wfc[f * 10 + c];
    }
    float m = logits[0];
#pragma unroll
    for (int c = 1; c < 10; ++c) m = fmaxf(m, logits[c]);
    float s = 0.0f;
#pragma unroll
    for (int c = 0; c < 10; ++c) { logits[c] = expf(logits[c] - m); s += logits[c]; }
    float invs = 1.0f / s;
#pragma unroll
    for (int c = 0; c < 10; ++c) out[g * 10 + c] = logits[c] * invs;
}

// ---------------------------------------------------------------------------
// Launcher
// ---------------------------------------------------------------------------
extern "C" void kernel_launch(void* const* d_in, const int* in_sizes, int n_in,
                              void* d_out, int out_size, void* d_ws, size_t ws_size,
                              hipStream_t stream) {
    const float* x    = (const float*)d_in[0];   // [n,128]
    const int*   ei   = (const int*)d_in[1];     // [2,E]
    const int*   bat  = (const int*)d_in[2];     // [n]
    const float* W1   = (const float*)d_in[3];   // [128,128]
    const float* b1   = (const float*)d_in[4];   // [128]
    const float* W2   = (const float*)d_in[5];   // [128,64]
    const float* b2   = (const float*)d_in[6];   // [64]
    const float* Wfc  = (const float*)d_in[7];   // [64,10]
    const float* bfc  = (const float*)d_in[8];   // [10]
    float* out = (float*)d_out;                  // [128,10]

    const int n = in_sizes[0] / 128;
    const int E = in_sizes[1] / 2;
    const int* src = ei;
    const int* dst = ei + E;

    // Workspace layout (floats). Layer-2 buffers alias layer-1 space.
    float* ws   = (float*)d_ws;
    float* h1   = ws;                          // [n,128] ; reused as h2 [n,64]
    float* out1 = h1 + (size_t)n * 128;        // [n,128] ; reused as out2 [n,64]
    float* deg  = out1 + (size_t)n * 128;      // [n]
    float* inv  = deg + n;                     // [n]
    float* sums = inv + n;                     // [128,64]
    float* cnts = sums + N_GRAPHS * 64;        // [128]
    float* h2   = h1;
    float* out2 = out1;

    const int T = 256;
    const int mtiles = (n + 15) / 16;

    // ---- degrees / normalization --------------------------------------
    k_fill<<<(n + T - 1) / T, T, 0, stream>>>(deg, 1.0f, n);                 // self loops
    k_fill<<<(N_GRAPHS * 64 + N_GRAPHS + T - 1) / T, T, 0, stream>>>(sums, 0.0f,
                                                                     N_GRAPHS * 64 + N_GRAPHS);
    k_deg_edges<<<(E + T - 1) / T, T, 0, stream>>>(dst, deg, E);
    k_inv_sqrt<<<(n + T - 1) / T, T, 0, stream>>>(deg, inv, n);

    // ---- layer 1: 128 -> 128 ------------------------------------------
    k_gemm_wmma<128, 128><<<mtiles, 8 * 32, 0, stream>>>(x, W1, h1, n);
    {
        size_t tot = (size_t)n * 128;
        k_self_init<128><<<(int)((tot + T - 1) / T), T, 0, stream>>>(h1, inv, out1, n);
        long long thr = (long long)E * 32;
        k_edge_scatter<128><<<(int)((thr + T - 1) / T), T, 0, stream>>>(src, dst, h1, inv, out1, E);
        k_bias_relu<128><<<(int)((tot + T - 1) / T), T, 0, stream>>>(out1, b1, tot);
    }

    // ---- layer 2: 128 -> 64 -------------------------------------------
    k_gemm_wmma<128, 64><<<mtiles, 4 * 32, 0, stream>>>(out1, W2, h2, n);
    {
        size_t tot = (size_t)n * 64;
        k_self_init<64><<<(int)((tot + T - 1) / T), T, 0, stream>>>(h2, inv, out2, n);
        long long thr = (long long)E * 16;
        k_edge_scatter<64><<<(int)((thr + T - 1) / T), T, 0, stream>>>(src, dst, h2, inv, out2, E);
        k_bias_relu<64><<<(int)((tot + T - 1) / T), T, 0, stream>>>(out2, b2, tot);
    }

    // ---- pooling + head ------------------------------------------------
    k_pool<<<(n + 511) / 512, 64, 0, stream>>>(out2, bat, sums, cnts, n);
    k_head<<<1, 128, 0, stream>>>(sums, cnts, Wfc, bfc, out);
}